// Walker_29351806501515
// MI455X (gfx1250) — compile-verified
//
#include <hip/hip_runtime.h>
#include <hip/hip_bf16.h>

// Non-backtracking random walk (Walker.sample_walks, start = arange(n)).
//
// One thread per walk. 32 sequential steps of L2-resident gathers; latency
// hidden by 3125 wave32 waves in flight. No matrix math exists in the
// reference, so no WMMA; the gfx1250-specific path used here is
// global_prefetch_b8 (via __builtin_prefetch) on the next step's coalesced
// `choices` row, the only memory stream independent of the serial chain.

__global__ __launch_bounds__(256) void Walker_29351806501515_kernel(
    const int* __restrict__ adj_nodes,   // [n * deg]
    const int* __restrict__ adj_offset,  // [n]
    const int* __restrict__ degrees,     // [n]
    const int* __restrict__ choices,     // [steps, n]
    float* __restrict__ walks,           // [steps+1, n]  (d_out part 1)
    float* __restrict__ walk_edges,      // [steps, n]    (d_out part 2)
    int n, int steps)
{
    int w = blockIdx.x * blockDim.x + threadIdx.x;
    if (w >= n) return;

    int cur  = w;     // start node = walk index (start_p = 1.0, arange(n))
    int prev = -1;    // step 0: backtrack test is all-false

    walks[w] = (float)cur;   // walks[0, w]

    for (int s = 0; s < steps; ++s) {
        // Prefetch next step's choice (coalesced stream, stride n*4 bytes);
        // lowers to global_prefetch_b8 on gfx1250.
        if (s + 1 < steps)
            __builtin_prefetch(&choices[(size_t)(s + 1) * n + w], 0, 1);

        // randint is in [0, 2^31): non-negative, so unsigned mod == jnp mod.
        unsigned ch  = (unsigned)choices[(size_t)s * n + w];
        unsigned deg = (unsigned)degrees[cur];
        int      off = adj_offset[cur];
        unsigned nbd = (deg > 1u) ? (deg - 1u) : 1u;   // max(deg-1, 1)

        unsigned eidx   = ch % deg;                    // modulo edge choice
        int      chosen = off + (int)eidx;
        int      nxt    = adj_nodes[chosen];

        if (nxt == prev) {                             // backtrack: rewrite edge
            unsigned aidx = (eidx + 1u + (ch % nbd)) % deg;
            chosen = off + (int)aidx;
            nxt    = adj_nodes[chosen];
        }

        // Coalesced stores; values < 2^24 so float conversion is exact.
        walks[(size_t)(s + 1) * n + w] = (float)nxt;
        walk_edges[(size_t)s * n + w]  = (float)chosen;

        prev = cur;
        cur  = nxt;
    }
}

extern "C" void kernel_launch(void* const* d_in, const int* in_sizes, int n_in,
                              void* d_out, int out_size, void* d_ws, size_t ws_size,
                              hipStream_t stream) {
    // setup_inputs() order: x(f32, unused), adj_nodes, adj_offset, degrees, choices
    const int* adj_nodes  = (const int*)d_in[1];
    const int* adj_offset = (const int*)d_in[2];
    const int* degrees    = (const int*)d_in[3];
    const int* choices    = (const int*)d_in[4];

    const int n     = in_sizes[3];           // 100000
    const int steps = in_sizes[4] / n;       // 32

    // Output tuple flattened in return order: walks [steps+1, n], edges [steps, n]
    float* walks      = (float*)d_out;
    float* walk_edges = walks + (size_t)(steps + 1) * n;

    const int block = 256;                   // 8 wave32 waves per block
    const int grid  = (n + block - 1) / block;

    Walker_29351806501515_kernel<<<grid, block, 0, stream>>>(
        adj_nodes, adj_offset, degrees, choices, walks, walk_edges, n, steps);
}